// ClusterSoftmax_14568529068676
// MI455X (gfx1250) — compile-verified
//
#include <hip/hip_runtime.h>
#include <hip/hip_bf16.h>

// CDNA5 / gfx1250: wave32, WMMA 16x16x4 f32.
typedef __attribute__((ext_vector_type(2))) float v2f;
typedef __attribute__((ext_vector_type(4))) float v4f;   // native vector: OK for NT builtins, b128 ld/st
typedef __attribute__((ext_vector_type(8))) float v8f;

#define BLOCK 256
#define WAVES_PER_BLOCK (BLOCK / 32)
#define GRID_REDUCE 2048
#define GRID_SCALE 2048

// 32-lane f32 reduction via one V_WMMA_F32_16X16X4_F32.
// A[m][0] = s_m (lanes 0-15), A[m][2] = s_{m+16} (lanes 16-31), other K = 0.
// B = all ones (4x16)  =>  D[m][n] = s_m + s_{m+16} = t_m for every column n.
// D layout: lane n (0-15) holds t_0..t_7, lane n (16-31) holds t_8..t_15 in its
// 8 accumulator VGPRs. Summing them gives a value UNIFORM within each 16-lane
// half: lanes 0-15 -> sum(t_0..t_7), lanes 16-31 -> sum(t_8..t_15).
__device__ __forceinline__ float wave_half_sum(float s) {
    v2f a;
    a.x = s;
    a.y = 0.0f;
    v2f b;
    b.x = 1.0f;
    b.y = 1.0f;
    v8f c = {};
    v8f d = __builtin_amdgcn_wmma_f32_16x16x4_f32(
        /*neg_a=*/false, a, /*neg_b=*/false, b,
        /*c_mod=*/(short)0, c, /*reuse_a=*/false, /*reuse_b=*/false);
    return ((d[0] + d[1]) + (d[2] + d[3])) + ((d[4] + d[5]) + (d[6] + d[7]));
}

__device__ __forceinline__ float masked_exp(float v) {
    return (v != 0.0f) ? __expf(v) : 0.0f;
}

// Block-level deterministic combine: each wave's two half-sums go to LDS
// (2 per wave), thread 0 adds the 2*WAVES_PER_BLOCK values in fixed order.
__device__ __forceinline__ void block_reduce_store(float s, float* dst) {
    __shared__ float smem[2 * WAVES_PER_BLOCK];
    const float u = wave_half_sum(s);  // EXEC all-ones here (no divergence)
    const int lane = threadIdx.x & 31;
    const int wave = threadIdx.x >> 5;
    if (lane == 0) smem[2 * wave + 0] = u;       // sum of t_0..t_7
    if (lane == 16) smem[2 * wave + 1] = u;      // sum of t_8..t_15
    __syncthreads();
    if (threadIdx.x == 0) {
        float t = 0.0f;
#pragma unroll
        for (int i = 0; i < 2 * WAVES_PER_BLOCK; ++i) t += smem[i];  // fixed order
        *dst = t;
    }
}

// Pass 1: per-block partial sums of e = (x != 0 ? exp(x) : 0).
// Regular (RT) loads on purpose: x (64 MB) should stay resident in the 192 MB
// L2 so pass 3 re-reads hit cache instead of HBM.
__global__ void __launch_bounds__(BLOCK)
partial_exp_sum_kernel(const float* __restrict__ x, float* __restrict__ partials, int n) {
    const v4f* __restrict__ x4 = reinterpret_cast<const v4f*>(x);
    const int n4 = n >> 2;
    const int stride = gridDim.x * blockDim.x;
    const int i0 = blockIdx.x * blockDim.x + threadIdx.x;

    // 2x unrolled grid-stride with independent accumulators: two b128 loads
    // in flight per iteration, two fp add chains.
    float s0 = 0.0f, s1 = 0.0f;
    int i = i0;
    for (; i + stride < n4; i += 2 * stride) {
        v4f va = x4[i];
        v4f vb = x4[i + stride];
        s0 += masked_exp(va.x) + masked_exp(va.y);
        s0 += masked_exp(va.z) + masked_exp(va.w);
        s1 += masked_exp(vb.x) + masked_exp(vb.y);
        s1 += masked_exp(vb.z) + masked_exp(vb.w);
    }
    for (; i < n4; i += stride) {
        v4f v = x4[i];
        s0 += masked_exp(v.x) + masked_exp(v.y);
        s0 += masked_exp(v.z) + masked_exp(v.w);
    }
    float s = s0 + s1;

    // Scalar tail (N % 4 == 0 in practice; robustness only).
    if (blockIdx.x == 0 && threadIdx.x == 0) {
        for (int t = n4 << 2; t < n; ++t) s += masked_exp(x[t]);
    }

    block_reduce_store(s, &partials[blockIdx.x]);
}

// Pass 2: single-block deterministic reduction of block partials -> denom.
__global__ void __launch_bounds__(BLOCK)
final_reduce_kernel(const float* __restrict__ partials, float* __restrict__ denom, int nparts) {
    float s = 0.0f;
    for (int i = threadIdx.x; i < nparts; i += BLOCK) s += partials[i];  // fixed order/thread
    block_reduce_store(s, denom);
}

// Pass 3: out = x * exp(x) / denom.  At x == 0 this is 0*1*inv == 0, exactly the
// reference's scatter-into-zeros semantics, so no mask needed.
// x loads: nontemporal (last use — hit L2 lines populated by pass 1, then let
// them age out).  out stores: nontemporal (write-only 64 MB stream; keep it
// from evicting anything useful in L2).
__global__ void __launch_bounds__(BLOCK)
scale_kernel(const float* __restrict__ x, const float* __restrict__ denom_p,
             float* __restrict__ out, int n) {
    const float inv = 1.0f / denom_p[0];
    const v4f* __restrict__ x4 = reinterpret_cast<const v4f*>(x);
    v4f* __restrict__ o4 = reinterpret_cast<v4f*>(out);
    const int n4 = n >> 2;
    const int stride = gridDim.x * blockDim.x;
    for (int i = blockIdx.x * blockDim.x + threadIdx.x; i < n4; i += stride) {
        v4f v = __builtin_nontemporal_load(&x4[i]);
        v4f r;
        r.x = v.x * __expf(v.x) * inv;
        r.y = v.y * __expf(v.y) * inv;
        r.z = v.z * __expf(v.z) * inv;
        r.w = v.w * __expf(v.w) * inv;
        __builtin_nontemporal_store(r, &o4[i]);
    }
    if (blockIdx.x == 0 && threadIdx.x == 0) {
        for (int i = n4 << 2; i < n; ++i) {
            float v = x[i];
            out[i] = v * __expf(v) * inv;
        }
    }
}

extern "C" void kernel_launch(void* const* d_in, const int* in_sizes, int n_in,
                              void* d_out, int out_size, void* d_ws, size_t ws_size,
                              hipStream_t stream) {
    const float* x = (const float*)d_in[0];
    float* out = (float*)d_out;
    const int n = in_sizes[0];

    float* partials = (float*)d_ws;          // GRID_REDUCE floats
    float* denom = partials + GRID_REDUCE;   // 1 float

    partial_exp_sum_kernel<<<GRID_REDUCE, BLOCK, 0, stream>>>(x, partials, n);
    final_reduce_kernel<<<1, BLOCK, 0, stream>>>(partials, denom, GRID_REDUCE);
    scale_kernel<<<GRID_SCALE, BLOCK, 0, stream>>>(x, denom, out, n);
}